// RandomizedStrideConvolution_14877766713728
// MI455X (gfx1250) — compile-verified
//
#include <hip/hip_runtime.h>
#include <hip/hip_bf16.h>

typedef float v2f __attribute__((ext_vector_type(2)));
typedef float v8f __attribute__((ext_vector_type(8)));

#define C_IO        32          // C_IN == C_OUT == 32 (one row = 128 B = 1 cacheline)
#define WAVES       8           // 256-thread block = 8 wave32
#define TPW         4           // tiles per wave (pipelined)
#define LDS_STRIDE  36          // dwords: 32 + pad -> 16B-aligned rows (144 B),
                                // conflict-free b64 A-fragment reads (36r mod 64 distinct)

// ---------------------------------------------------------------------------
// out[n][c] = bias[c]  (scatter kernel then atomically accumulates on top)
// ---------------------------------------------------------------------------
__global__ __launch_bounds__(256) void init_out_kernel(float* __restrict__ out,
                                                       const float* __restrict__ bias,
                                                       int total) {
    int tid = blockIdx.x * blockDim.x + threadIdx.x;
    if (tid < total) out[tid] = bias[tid & (C_IO - 1)];
}

// ---------------------------------------------------------------------------
// Async gather -> WMMA f32 16x16x4 GEMM -> atomic scatter-add.
// One wave owns a 16-row tile: out_tile[16x32] = in_tile[16x32] * W[32x32].
// Double-buffered LDS; next tile's gather is in flight (ASYNCcnt) while the
// current tile runs through the matrix pipe and the L2 atomics.
// ---------------------------------------------------------------------------
__global__ __launch_bounds__(256) void spconv_wmma_kernel(
    const float* __restrict__ feats,   // [N_IN, 32]
    const float* __restrict__ weight,  // [K, 1, 32, 32]
    const int*   __restrict__ rin,     // [K, R]
    const int*   __restrict__ rout,    // [K, R]
    float*       __restrict__ out,     // [N_OUT, 32]
    int R, int tiles_per_k)
{
    __shared__ float smem[WAVES * 2 * 16 * LDS_STRIDE];   // 36 KB, 2 buffers/wave

    const int lane = threadIdx.x & 31;
    const int wave = threadIdx.x >> 5;
    const int half = lane >> 4;       // 0: lanes 0-15, 1: lanes 16-31
    const int l16  = lane & 15;
    const int k    = blockIdx.y;

    float* sm = smem + wave * (2 * 16 * LDS_STRIDE);

    const int tile0  = (blockIdx.x * WAVES + wave) * TPW;
    int ntiles = tiles_per_k - tile0;
    if (ntiles <= 0) return;                 // wave-uniform; no block barriers exist
    if (ntiles > TPW) ntiles = TPW;

    // --- Preload B fragments for this tap (32 VGPRs per lane). ---
    // ISA 4x16 f32 B layout: VGPR v, half-wave h holds row K = 4j + v + 2h,
    // column N = nb*16 + l16.
    const float* W = weight + k * (C_IO * C_IO);
    float wreg[2][8][2];
#pragma unroll
    for (int j = 0; j < 8; ++j) {
#pragma unroll
        for (int v = 0; v < 2; ++v) {
            int cin = 4 * j + v + 2 * half;
            wreg[0][j][v] = W[cin * C_IO + l16];
            wreg[1][j][v] = W[cin * C_IO + 16 + l16];
        }
    }

    const int* rin_k  = rin  + k * R;
    const int* rout_k = rout + k * R;

    // Per-lane index load for a tile (lanes 0-15 own rows 0-15; hi half duplicated).
    auto load_idx = [&](int tile) -> int {
        int myrow = tile * 16 + l16;
        return (myrow < R) ? rin_k[myrow] : 0;   // clamp tail to a safe row
    };

    // Async-gather one 16x32 f32 tile (2 KB) into an LDS buffer:
    // 4 x global_load_async_to_lds_b128, each lane moves 16 B of one row.
    auto issue_gather = [&](float* sm_buf, int gidx) {
        uint32_t lds_base = (uint32_t)(size_t)sm_buf;   // low 32b of flat = LDS offset
        int seg = lane & 7;                              // 16B segment within the row
#pragma unroll
        for (int q = 0; q < 4; ++q) {
            int row = q * 4 + (lane >> 3);               // 4 rows per instruction
            int idx = __shfl(gidx, row, 32);
            const float* gp = feats + (size_t)idx * C_IO + seg * 4;
            uint32_t lo = lds_base + row * (LDS_STRIDE * 4) + seg * 16;
            asm volatile("global_load_async_to_lds_b128 %0, %1, off"
                         :: "v"(lo), "v"(gp) : "memory");
        }
    };

    // Prime the pipeline.
    issue_gather(sm, load_idx(tile0));

    for (int t = 0; t < ntiles; ++t) {
        int row0 = (tile0 + t) * 16;

        // Wait for this tile's async gather to land in LDS.
        asm volatile("s_wait_asynccnt 0x0" ::: "memory");
        __builtin_amdgcn_wave_barrier();

        // Kick off the next tile's gather into the other buffer (overlaps
        // with the WMMA chain + scatter atomics below).
        if (t + 1 < ntiles)
            issue_gather(sm + ((t + 1) & 1) * (16 * LDS_STRIDE),
                         load_idx(tile0 + t + 1));

        // --- 16 x v_wmma_f32_16x16x4_f32: accumulate K=32 in steps of 4. ---
        // A 16x4 f32 layout: VGPR v, half-wave h holds A[M=l16][K = 4j + v + 2h].
        const float* arow = sm + (t & 1) * (16 * LDS_STRIDE)
                               + l16 * LDS_STRIDE + 2 * half;
        v8f acc0 = {};
        v8f acc1 = {};
#pragma unroll
        for (int j = 0; j < 8; ++j) {
            v2f a;
            a.x = arow[4 * j + 0];
            a.y = arow[4 * j + 1];
            v2f b0; b0.x = wreg[0][j][0]; b0.y = wreg[0][j][1];
            v2f b1; b1.x = wreg[1][j][0]; b1.y = wreg[1][j][1];
            acc0 = __builtin_amdgcn_wmma_f32_16x16x4_f32(
                false, a, false, b0, (short)0, acc0, false, false);
            acc1 = __builtin_amdgcn_wmma_f32_16x16x4_f32(
                false, a, false, b1, (short)0, acc1, false, false);
        }

        // --- Scatter: C/D layout VGPR r, half h -> M = 8h + r, N = l16. ---
#pragma unroll
        for (int r = 0; r < 8; ++r) {
            int rr = row0 + half * 8 + r;
            if (rr < R) {
                int orow   = rout_k[rr];
                float* dst = out + orow * C_IO + l16;
                atomicAdd(dst,      acc0[r]);
                atomicAdd(dst + 16, acc1[r]);
            }
        }
    }
}

// ---------------------------------------------------------------------------
extern "C" void kernel_launch(void* const* d_in, const int* in_sizes, int n_in,
                              void* d_out, int out_size, void* d_ws, size_t ws_size,
                              hipStream_t stream) {
    const float* feats  = (const float*)d_in[0];
    const float* weight = (const float*)d_in[1];
    const float* bias   = (const float*)d_in[2];
    const int*   rin    = (const int*)d_in[3];
    const int*   rout   = (const int*)d_in[4];
    float*       out    = (float*)d_out;

    const int K = in_sizes[1] / (C_IO * C_IO);   // weight flat = K*1*32*32
    const int R = in_sizes[3] / K;               // rules_in flat = K*R

    // 1) out = bias (broadcast)
    int total  = out_size;
    int iblks  = (total + 255) / 256;
    init_out_kernel<<<iblks, 256, 0, stream>>>(out, bias, total);

    // 2) async gather -> WMMA -> scatter-add
    int tiles_per_k    = (R + 15) / 16;
    int tiles_per_blk  = WAVES * TPW;
    int blocks_per_k   = (tiles_per_k + tiles_per_blk - 1) / tiles_per_blk;
    dim3 grid(blocks_per_k, K);
    spconv_wmma_kernel<<<grid, 256, 0, stream>>>(feats, weight, rin, rout, out,
                                                 R, tiles_per_k);
}